// RPN_44074954391689
// MI455X (gfx1250) — compile-verified
//
#include <hip/hip_runtime.h>
#include <hip/hip_bf16.h>
#include <stdint.h>

#define BB 4
#define MM 5000
#define KK 2000
#define NMS_THRESH 0.7f
#define NCB ((MM + 31) / 32)   // 157 column blocks of 32 (wave32)
#define SORT_N 8192            // pow2 >= MM

// ---------------------------------------------------------------------------
// CDNA5 helpers: LDS byte-offset for async DMA, async global->LDS copy
// (tracked by ASYNCcnt), and the split-counter wait.
// ---------------------------------------------------------------------------
__device__ __forceinline__ unsigned lds_off(void* p) {
  // generic -> addrspace(3); AS3 pointers are 32-bit LDS byte offsets
  return (unsigned)(unsigned long long)(__attribute__((address_space(3))) void*)p;
}

// GVS form: global_load_async_to_lds_b128 vdst(LDS off), vaddr(32b off), saddr(64b base)
__device__ __forceinline__ void async_load_b128(unsigned lds_byte_off,
                                                unsigned voff,
                                                const void* base) {
  asm volatile("global_load_async_to_lds_b128 %0, %1, %2"
               :
               : "v"(lds_byte_off), "v"(voff),
                 "s"((unsigned long long)(uintptr_t)base)
               : "memory");
}

__device__ __forceinline__ void wait_asynccnt0() {
  asm volatile("s_wait_asynccnt 0x0" ::: "memory");
}

// ---------------------------------------------------------------------------
// Kernel 0: zero the output (rois padding + the two scalar losses)
// ---------------------------------------------------------------------------
__global__ void nms_zero_kernel(float* __restrict__ out, int n) {
  int i = blockIdx.x * blockDim.x + threadIdx.x;
  if (i < n) out[i] = 0.0f;
}

// ---------------------------------------------------------------------------
// Kernel 1: per-batch descending argsort by score (bitonic, LDS-resident),
// then gather boxes into score order. Keys: (score_bits<<32)|index, u64.
// Positive-float bit patterns are order-preserving, pads are key=0.
// ---------------------------------------------------------------------------
__global__ __launch_bounds__(1024)
void nms_sort_kernel(const float* __restrict__ boxes,
                     const float* __restrict__ scores,
                     float* __restrict__ sortedBoxes) {
  __shared__ unsigned long long key[SORT_N];  // 64 KB LDS (<= 320 KB/WGP)
  const int b   = blockIdx.x;
  const int tid = threadIdx.x;

  for (int i = tid; i < SORT_N; i += 1024) {
    if (i < MM) {
      unsigned sb = __float_as_uint(scores[b * MM + i]);
      key[i] = ((unsigned long long)sb << 32) | (unsigned)i;
    } else {
      key[i] = 0ull;
    }
  }
  __syncthreads();

  // ascending bitonic sort; descending rank r lives at position SORT_N-1-r
  for (int k = 2; k <= SORT_N; k <<= 1) {
    for (int j = k >> 1; j > 0; j >>= 1) {
      for (int i = tid; i < SORT_N; i += 1024) {
        int ixj = i ^ j;
        if (ixj > i) {
          unsigned long long a = key[i];
          unsigned long long c = key[ixj];
          bool asc = ((i & k) == 0);
          if ((a > c) == asc) { key[i] = c; key[ixj] = a; }
        }
      }
      __syncthreads();
    }
  }

  const float4* bsrc = (const float4*)boxes + (size_t)b * MM;
  float4*       bdst = (float4*)sortedBoxes + (size_t)b * MM;
  for (int r = tid; r < MM; r += 1024) {
    unsigned src = (unsigned)(key[SORT_N - 1 - r] & 0xFFFFFFFFull);
    bdst[r] = bsrc[src];
  }
}

// ---------------------------------------------------------------------------
// Kernel 2: IoU suppression bitmask. Each block: 256 rows x one 32-column
// block. Column boxes staged into LDS via CDNA5 async global->LDS DMA
// (wave 0 issues 32 x b128 DMAs, waits ASYNCcnt, workgroup barrier publishes).
// Triangular early-out: bits need col > row, so (rowBase >= colBase+31)
// tiles are identically zero -> skip DMA + IoU math, store 0.
// ---------------------------------------------------------------------------
__global__ __launch_bounds__(256)
void nms_mask_kernel(const float* __restrict__ sortedBoxes,
                     unsigned* __restrict__ mask) {
  __shared__ float4 cb[32];
  const int b       = blockIdx.z;
  const int cbi     = blockIdx.x;
  const int tid     = threadIdx.x;
  const int colBase = cbi * 32;
  const int rowBase = blockIdx.y * 256;
  const float4* sb  = (const float4*)sortedBoxes + (size_t)b * MM;

  const bool blockDead = rowBase >= colBase + 31;  // whole tile below diagonal

  if (!blockDead && tid < 32) {  // wave 0: 32 lanes -> 32 x 16B async DMA
    int col = colBase + tid;
    if (col >= MM) col = MM - 1;  // dup tail; masked out below by col<MM
    async_load_b128(lds_off(&cb[tid]), (unsigned)(col * 16), (const void*)sb);
    wait_asynccnt0();
  }
  __syncthreads();

  const int row = rowBase + tid;
  if (row >= MM) return;

  unsigned* dst = mask + ((size_t)b * MM + row) * NCB + cbi;
  if (blockDead || row >= colBase + 31) {  // no col in this word can exceed row
    *dst = 0u;
    return;
  }

  float4 r = sb[row];
  float rArea = (r.z - r.x + 1.0f) * (r.w - r.y + 1.0f);
  unsigned bits = 0u;
#pragma unroll
  for (int j = 0; j < 32; ++j) {
    int col = colBase + j;
    float4 c = cb[j];
    float xx1 = fmaxf(r.x, c.x);
    float yy1 = fmaxf(r.y, c.y);
    float xx2 = fminf(r.z, c.z);
    float yy2 = fminf(r.w, c.w);
    float iw = fmaxf(xx2 - xx1 + 1.0f, 0.0f);
    float ih = fmaxf(yy2 - yy1 + 1.0f, 0.0f);
    float inter = iw * ih;
    float cArea = (c.z - c.x + 1.0f) * (c.w - c.y + 1.0f);
    float iou = inter / (rArea + cArea - inter);
    if ((col > row) && (col < MM) && (iou > NMS_THRESH)) bits |= (1u << j);
  }
  *dst = bits;
}

// ---------------------------------------------------------------------------
// Kernel 3: serial keep-scan, one wave32 per batch, word-at-a-time.
// Suppression bitset (157 words) lives in LDS. ctz jumps straight to the
// next live row, so suppressed rows cost ~0 iterations; early exit at K.
// All branching is wave-uniform. Writes rois rows [b, x1, y1, x2, y2].
// ---------------------------------------------------------------------------
__global__ __launch_bounds__(32)
void nms_keep_kernel(const float* __restrict__ sortedBoxes,
                     const unsigned* __restrict__ mask,
                     float* __restrict__ out) {
  __shared__ unsigned removed[NCB];
  const int b    = blockIdx.x;
  const int lane = threadIdx.x;

  for (int w = lane; w < NCB; w += 32) removed[w] = 0u;
  __syncthreads();

  const float4* sb = (const float4*)sortedBoxes + (size_t)b * MM;
  int kept = 0;
  for (int cbw = 0; cbw < NCB && kept < KK; ++cbw) {
    const int base  = cbw * 32;
    const int nbits = (MM - base < 32) ? (MM - base) : 32;
    const unsigned validMask = (nbits == 32) ? 0xFFFFFFFFu : ((1u << nbits) - 1u);
    unsigned live = ~removed[cbw] & validMask;   // wave-uniform LDS broadcast
    while (live != 0u && kept < KK) {
      const int i = __builtin_ctz(live);         // next unsuppressed row
      const int r = base + i;
      if (lane == 0) {
        float4 bx = sb[r];
        float* o = out + ((size_t)b * KK + kept) * 5;
        o[0] = (float)b; o[1] = bx.x; o[2] = bx.y; o[3] = bx.z; o[4] = bx.w;
      }
      kept++;
      // fold this kept row's suppression mask into the bitset (~5 words/lane)
      const unsigned* mrow = mask + ((size_t)b * MM + r) * NCB;
      for (int w = lane; w < NCB; w += 32) removed[w] |= mrow[w];
      __syncthreads();  // single wave: orders DS writes before the refresh
      live &= ~(1u << i);
      live &= ~removed[cbw];  // this keep may suppress later rows in same word
    }
  }
}

// ---------------------------------------------------------------------------
extern "C" void kernel_launch(void* const* d_in, const int* in_sizes, int n_in,
                              void* d_out, int out_size, void* d_ws, size_t ws_size,
                              hipStream_t stream) {
  (void)in_sizes; (void)n_in; (void)ws_size;
  const float* boxes  = (const float*)d_in[0];  // (B, M, 4) f32
  const float* scores = (const float*)d_in[1];  // (B, M)    f32
  float* out = (float*)d_out;                   // 40000 rois + 2 scalar losses

  char* ws = (char*)d_ws;
  float*    sortedBoxes = (float*)ws;                  // B*M*4 f32  = 320 000 B
  unsigned* mask        = (unsigned*)(ws + (1 << 19)); // +512KB; B*M*NCB u32 ~ 12.56 MB

  nms_zero_kernel<<<(out_size + 255) / 256, 256, 0, stream>>>(out, out_size);
  nms_sort_kernel<<<BB, 1024, 0, stream>>>(boxes, scores, sortedBoxes);
  dim3 mgrid(NCB, (MM + 255) / 256, BB);
  nms_mask_kernel<<<mgrid, 256, 0, stream>>>(sortedBoxes, mask);
  nms_keep_kernel<<<BB, 32, 0, stream>>>(sortedBoxes, mask, out);
}